// PANLoss_67379446940211
// MI455X (gfx1250) — compile-verified
//
#include <hip/hip_runtime.h>
#include <hip/hip_bf16.h>
#include <math.h>

// ---------------- problem constants (from reference) ----------------
#define B_    8
#define CH    6          // 2 + FEATURE_DIM
#define HLO   160
#define WLO   160
#define HHI   640
#define WHI   640
#define NPIX  (HHI*WHI)   // 409600
#define NCELL (HLO*WLO)   // 25600
#define KLAB  8
#define DFEA  4
#define EPS_DICE 0.001f
#define DELTA_V  0.5f
#define DELTA_D  1.5f

typedef __attribute__((ext_vector_type(2))) float v2f;
typedef __attribute__((ext_vector_type(8))) float v8f;
typedef __attribute__((ext_vector_type(4))) unsigned int u32x4;
typedef __attribute__((ext_vector_type(8))) int i32x8;
typedef __attribute__((ext_vector_type(4))) int i32x4;

#if __has_builtin(__builtin_amdgcn_tensor_load_to_lds) && __has_builtin(__builtin_amdgcn_s_wait_tensorcnt)
#define HAVE_TDM 1
#endif

// ---------------- workspace layout ----------------
struct Ws {
    unsigned pos_num[B_];
    unsigned neg_cnt[B_];
    float    sum_e[B_][DFEA][KLAB];
    unsigned cnt_k[B_][KLAB];
    unsigned cnt_i[B_][KLAB];
    unsigned prefix[B_];
    unsigned remaining[B_];
    unsigned fallback[B_];
    float    thr[B_];
    float    emb_mean[B_][DFEA][KLAB];
    float    dice[B_][6];           // a1,b1,c1,a2,b2,c2
    float    sum_val[B_][KLAB];
    unsigned hist[B_][256];
    unsigned negmult[B_][NCELL];
};

// ---------------- helpers ----------------
__device__ __forceinline__ unsigned f2key(float f) {
    unsigned u = __float_as_uint(f);
    return (u & 0x80000000u) ? ~u : (u | 0x80000000u);   // monotone: bigger float -> bigger key
}
__device__ __forceinline__ float key2f(unsigned k) {
    return (k & 0x80000000u) ? __uint_as_float(k & 0x7FFFFFFFu) : __uint_as_float(~k);
}
__device__ __forceinline__ float sigm(float x) { return 1.f / (1.f + __expf(-x)); }

// ---------------- TDM staging: one 160x6 f32 tile (maps strip, all channels) -> LDS ----------------
// D# group0: count=1, lds_addr, global_addr (57b), type=2 ("image")
// D# group1: data_size=4B, tensor_dim0=NCELL, tensor_dim1=CH, tile_dim0=WLO, tile_dim1=CH,
//            tensor_dim0_stride=NCELL (channel stride in elements)
#ifdef HAVE_TDM
__device__ __forceinline__ void tdm_stage_maps_strip(const float* gsrc, void* lds_dst) {
    unsigned long long ga = (unsigned long long)(uintptr_t)gsrc;
    unsigned lds_off = (unsigned)(uintptr_t)lds_dst;   // addrspace(3) byte offset
    u32x4 g0;
    g0[0] = 1u;                                           // count=1, user descriptor
    g0[1] = lds_off;                                      // lds_addr
    g0[2] = (unsigned)(ga & 0xFFFFFFFFull);               // global_addr[31:0]
    g0[3] = (unsigned)((ga >> 32) & 0x01FFFFFFull)        // global_addr[56:32]
          | (2u << 30);                                   // type=2
    i32x8 g1;
    g1[0] = (int)(2u << 16);                              // data_size = 4 bytes
    g1[1] = (int)(((unsigned)NCELL & 0xFFFFu) << 16);     // tensor_dim0 lo16 -> [31:16]
    g1[2] = (int)((((unsigned)NCELL >> 16) & 0xFFFFu)     // tensor_dim0 hi16
          | ((unsigned)CH << 16));                        // tensor_dim1 lo16
    g1[3] = (int)((unsigned)WLO << 16);                   // tensor_dim1 hi16 | tile_dim0
    g1[4] = (int)(unsigned)CH;                            // tile_dim1 | tile_dim2(0)
    g1[5] = (int)(unsigned)NCELL;                         // tensor_dim0_stride lo32
    g1[6] = 0;                                            // stride0 hi16 | stride1 lo16
    g1[7] = 0;                                            // stride1 hi32
    i32x4 z4 = {0, 0, 0, 0};
#if defined(__clang_major__) && (__clang_major__ >= 23)
    i32x8 z8 = {0, 0, 0, 0, 0, 0, 0, 0};
    __builtin_amdgcn_tensor_load_to_lds(g0, g1, z4, z4, z8, 0);
#else
    __builtin_amdgcn_tensor_load_to_lds(g0, g1, z4, z4, 0);
#endif
}
#endif

// Stage one low-res row (all CH channels) of maps into LDS, then barrier.
__device__ __forceinline__ void stage_strip(const float* mb, int strip, float (*s_maps)[WLO]) {
#ifdef HAVE_TDM
    if (threadIdx.x < 32) {               // wave 0 only issues; other waves branch over it
        tdm_stage_maps_strip(mb + (size_t)strip * WLO, &s_maps[0][0]);
        __builtin_amdgcn_s_wait_tensorcnt(0);
    }
#else
    for (int i = threadIdx.x; i < CH * WLO; i += blockDim.x)
        ((float*)s_maps)[i] = mb[(size_t)(i / WLO) * NCELL + (size_t)strip * WLO + (i % WLO)];
#endif
    __syncthreads();
}

// ---------------- kernel 0: zero scratch + output ----------------
__global__ void k_init(Ws* ws, float* out) {
    size_t n = sizeof(Ws) / 4;
    unsigned* p = (unsigned*)ws;
    for (size_t i = (size_t)blockIdx.x * blockDim.x + threadIdx.x; i < n;
         i += (size_t)gridDim.x * blockDim.x) p[i] = 0u;
    if (blockIdx.x == 0 && threadIdx.x == 0) out[0] = 0.f;
}

// ---------------- kernel 1: per-cell negative multiplicity ----------------
__global__ void k_negmult(const float* __restrict__ gt_texts, Ws* ws) {
    int b = blockIdx.y;
    int c = blockIdx.x * blockDim.x + threadIdx.x;
    if (c >= NCELL) return;
    int ly = c / WLO, lx = c % WLO;
    const float* g = gt_texts + (size_t)b * NPIX;
    unsigned cnt = 0;
    #pragma unroll
    for (int dy = 0; dy < 4; ++dy) {
        const float* row = g + (size_t)(ly * 4 + dy) * WHI + lx * 4;
        #pragma unroll
        for (int dx = 0; dx < 4; ++dx) cnt += (row[dx] <= 0.5f) ? 1u : 0u;
    }
    ws->negmult[b][c] = cnt;
}

// ---------------- kernel 2: streaming pass 1 (counts + one-hot einsum sums) ----------------
// One block per low-res strip (4 full-res rows); maps strip staged in LDS via TDM.
__global__ void k_pass1(const float* __restrict__ maps, const float* __restrict__ gt_texts,
                        const float* __restrict__ gt_kernels, const int* __restrict__ gt_instance,
                        const float* __restrict__ tmask, Ws* ws) {
    int strip = blockIdx.x;                 // 0..HLO-1
    int b     = blockIdx.y;
    __shared__ float s_maps[CH][WLO];
    __shared__ unsigned s_pos, s_neg, s_cntk[KLAB], s_cnti[KLAB];
    __shared__ float s_sume[DFEA][KLAB];
    if (threadIdx.x == 0) { s_pos = 0u; s_neg = 0u; }
    if (threadIdx.x < KLAB) { s_cntk[threadIdx.x] = 0u; s_cnti[threadIdx.x] = 0u; }
    if (threadIdx.x < DFEA * KLAB) ((float*)s_sume)[threadIdx.x] = 0.f;

    const float* mb = maps + (size_t)b * CH * NCELL;
    stage_strip(mb, strip, s_maps);         // includes __syncthreads()

    const float* gt = gt_texts    + (size_t)b * NPIX;
    const float* gk = gt_kernels  + (size_t)b * NPIX;
    const int*   gi = gt_instance + (size_t)b * NPIX;
    const float* tm = tmask       + (size_t)b * NPIX;

    int base = strip * 4 * WHI;             // first pixel of this strip
    for (int p = threadIdx.x; p < 4 * WHI; p += blockDim.x) {
        int dy = p / WHI, x = p - dy * WHI;
        int n  = base + dy * WHI + x;
        int lx = x >> 2;
        __builtin_prefetch(&gt[n + (int)blockDim.x], 0, 1);   // global_prefetch_b8
        float g = gt[n], t = tm[n];
        bool pos = g > 0.5f;
        if (pos && t > 0.5f) atomicAdd(&s_pos, 1u);
        if (!pos)            atomicAdd(&s_neg, 1u);
        int inst = gi[n]; inst = inst < 0 ? 0 : (inst >= KLAB ? KLAB - 1 : inst);
        bool tb = t > 0.5f;
        int labk = (tb && gk[n] > 0.5f) ? inst : 0;   // kernel-masked label (0 = background bucket)
        int labi = tb ? inst : 0;
        atomicAdd(&s_cntk[labk], 1u);
        atomicAdd(&s_cnti[labi], 1u);
        #pragma unroll
        for (int d = 0; d < DFEA; ++d)
            atomicAdd(&s_sume[d][labk], s_maps[2 + d][lx]);
    }
    __syncthreads();
    if (threadIdx.x == 0) { atomicAdd(&ws->pos_num[b], s_pos); atomicAdd(&ws->neg_cnt[b], s_neg); }
    if (threadIdx.x < KLAB) {
        atomicAdd(&ws->cnt_k[b][threadIdx.x], s_cntk[threadIdx.x]);
        atomicAdd(&ws->cnt_i[b][threadIdx.x], s_cnti[threadIdx.x]);
    }
    if (threadIdx.x < DFEA * KLAB)
        atomicAdd(&((float*)ws->sum_e[b])[threadIdx.x], ((float*)s_sume)[threadIdx.x]);
}

// ---------------- kernel 3: per-batch prep (neg_num, means, select init) ----------------
__global__ void k_prep(Ws* ws) {
    int b = threadIdx.x;
    if (b >= B_) return;
    unsigned pos = ws->pos_num[b], neg = ws->neg_cnt[b];
    unsigned long long nn3 = (unsigned long long)pos * 3ull;
    unsigned neg_num = (unsigned)(nn3 < (unsigned long long)neg ? nn3 : neg);
    ws->remaining[b] = neg_num;
    ws->prefix[b]    = 0u;
    ws->fallback[b]  = (pos == 0u || neg_num == 0u) ? 1u : 0u;
    ws->thr[b]       = -3.402823466e+38f;
    for (int k = 0; k < KLAB; ++k) {
        float c = (float)ws->cnt_k[b][k]; if (c < 1.f) c = 1.f;
        for (int d = 0; d < DFEA; ++d) {
            float m = ws->sum_e[b][d][k] / c;
            ws->emb_mean[b][d][k] = (k == 0) ? 0.f : m;   // nz masking
        }
    }
}

// ---------------- kernel 4: radix-select histogram pass (over low-res cells) ----------------
__global__ void k_hist(const float* __restrict__ maps, Ws* ws, int pass) {
    int b = blockIdx.y;
    __shared__ unsigned sh[256];
    sh[threadIdx.x] = 0u;                       // blockDim.x == 256
    __syncthreads();
    unsigned prefix = ws->prefix[b];
    unsigned mask   = (pass == 0) ? 0u : (0xFFFFFFFFu << (32 - 8 * pass));
    int shift = 24 - 8 * pass;
    const float* sc = maps + (size_t)b * CH * NCELL;       // channel 0 (texts)
    for (int c = blockIdx.x * blockDim.x + threadIdx.x; c < NCELL; c += gridDim.x * blockDim.x) {
        unsigned mult = ws->negmult[b][c];
        if (mult) {
            unsigned key = f2key(sc[c]);
            if ((key & mask) == prefix) atomicAdd(&sh[(key >> shift) & 0xFFu], mult);
        }
    }
    __syncthreads();
    if (sh[threadIdx.x]) atomicAdd(&ws->hist[b][threadIdx.x], sh[threadIdx.x]);
}

// ---------------- kernel 5: pick bin, update prefix/remaining, reset hist ----------------
__global__ void k_select(Ws* ws, int pass) {
    int b = blockIdx.x;
    __shared__ unsigned sh[256];
    sh[threadIdx.x] = ws->hist[b][threadIdx.x];
    __syncthreads();
    ws->hist[b][threadIdx.x] = 0u;              // zero for next pass
    if (threadIdx.x == 0) {
        unsigned rem = ws->remaining[b];
        if (rem > 0u) {
            unsigned cum = 0u; int chosen = 0;
            for (int bin = 255; bin >= 0; --bin) {
                unsigned nc = cum + sh[bin];
                if (nc >= rem) { chosen = bin; break; }
                cum = nc;
            }
            ws->remaining[b] = rem - cum;
            unsigned pref = ws->prefix[b] | ((unsigned)chosen << (24 - 8 * pass));
            ws->prefix[b] = pref;
            if (pass == 3) ws->thr[b] = key2f(pref);   // exact k-th largest neg score
        }
    }
}

// ---------------- kernel 6: streaming pass 2 (dice sums + per-label val sums) ----------------
__global__ void k_pass2(const float* __restrict__ maps, const float* __restrict__ gt_texts,
                        const float* __restrict__ gt_kernels, const int* __restrict__ gt_instance,
                        const float* __restrict__ tmask, Ws* ws) {
    int strip = blockIdx.x;
    int b     = blockIdx.y;
    __shared__ float s_maps[CH][WLO];
    __shared__ float s_acc[6];
    __shared__ float s_val[KLAB];
    __shared__ float s_mean[DFEA][KLAB];
    if (threadIdx.x < 6)    s_acc[threadIdx.x] = 0.f;
    if (threadIdx.x < KLAB) s_val[threadIdx.x] = 0.f;
    if (threadIdx.x < DFEA * KLAB)
        ((float*)s_mean)[threadIdx.x] = ((float*)ws->emb_mean[b])[threadIdx.x];

    const float* mb = maps + (size_t)b * CH * NCELL;
    stage_strip(mb, strip, s_maps);         // includes __syncthreads()

    float thr   = ws->thr[b];
    unsigned fb = ws->fallback[b];
    const float* gt = gt_texts    + (size_t)b * NPIX;
    const float* gk = gt_kernels  + (size_t)b * NPIX;
    const int*   gi = gt_instance + (size_t)b * NPIX;
    const float* tm = tmask       + (size_t)b * NPIX;

    float a1 = 0.f, b1 = 0.f, c1 = 0.f, a2 = 0.f, b2 = 0.f, c2 = 0.f;
    int base = strip * 4 * WHI;
    for (int p = threadIdx.x; p < 4 * WHI; p += blockDim.x) {
        int dy = p / WHI, x = p - dy * WHI;
        int n  = base + dy * WHI + x;
        int lx = x >> 2;
        __builtin_prefetch(&tm[n + (int)blockDim.x], 0, 1);
        float score = s_maps[0][lx];               // texts (ch0)
        float kv    = s_maps[1][lx];               // kernels (ch1)
        float g = gt[n], t = tm[n];
        float sel = fb ? t : ((((score >= thr) || (g > 0.5f)) && (t > 0.5f)) ? 1.f : 0.f);
        float sp = sigm(score);
        float p_ = sp * sel, t_ = g * sel;
        a1 += p_ * t_; b1 += p_ * p_; c1 += t_ * t_;
        float selk = (sp > 0.5f ? 1.f : 0.f) * t;
        float pk = sigm(kv) * selk, tk = gk[n] * selk;
        a2 += pk * tk; b2 += pk * pk; c2 += tk * tk;
        // embedding aggregation term
        int inst = gi[n]; inst = inst < 0 ? 0 : (inst >= KLAB ? KLAB - 1 : inst);
        int labi = (t > 0.5f) ? inst : 0;
        float sqd = 0.f;
        #pragma unroll
        for (int d = 0; d < DFEA; ++d) {
            float df = s_maps[2 + d][lx] - s_mean[d][labi];
            sqd += df * df;
        }
        float dist = sqrtf(fmaxf(sqd, 1e-12f));
        float dv = fmaxf(dist - DELTA_V, 0.f);
        atomicAdd(&s_val[labi], logf(dv * dv + 1.f));
    }
    atomicAdd(&s_acc[0], a1); atomicAdd(&s_acc[1], b1); atomicAdd(&s_acc[2], c1);
    atomicAdd(&s_acc[3], a2); atomicAdd(&s_acc[4], b2); atomicAdd(&s_acc[5], c2);
    __syncthreads();
    if (threadIdx.x < 6)    atomicAdd(&ws->dice[b][threadIdx.x], s_acc[threadIdx.x]);
    if (threadIdx.x < KLAB) atomicAdd(&ws->sum_val[b][threadIdx.x], s_val[threadIdx.x]);
}

// ---------------- kernel 7: per-batch finale — WMMA Gram matrix + scalar epilogue ----------------
// One wave (32 lanes) per image. Gram = M^T M, M = 4x8 f32 instance means,
// computed with a single exact V_WMMA_F32_16X16X4_F32 (K = DFEA = 4).
__global__ void k_final(Ws* ws, float* __restrict__ out) {
    int b = blockIdx.x;
    int lane = threadIdx.x;                 // 0..31, wave32
    __shared__ float gram[16][16];
    __shared__ float mean[DFEA][KLAB];
    if (lane < DFEA * KLAB) ((float*)mean)[lane] = ((float*)ws->emb_mean[b])[lane];
    __syncthreads();

    // A (16x4 f32): lanes 0-15 hold K=0,1 ; lanes 16-31 hold K=2,3 (ISA 7.12.2)
    // A[m=label r][k=feature] = mean[k][r]; B (4x16) mirrors it -> identical lane values.
    int r  = lane & 15;
    int kb = (lane < 16) ? 0 : 2;
    v2f a;
    a.x = (r < KLAB) ? mean[kb][r]     : 0.f;
    a.y = (r < KLAB) ? mean[kb + 1][r] : 0.f;
    v2f bm = a;
    v8f c = {0.f, 0.f, 0.f, 0.f, 0.f, 0.f, 0.f, 0.f};
    c = __builtin_amdgcn_wmma_f32_16x16x4_f32(false, a, false, bm, (short)0, c, false, false);
    #pragma unroll
    for (int v = 0; v < 8; ++v) {
        int m = v + ((lane < 16) ? 0 : 8);   // C/D layout: vgpr v = row M=v / v+8
        gram[m][r] = c[v];
    }
    __syncthreads();

    if (lane == 0) {
        unsigned cntk[KLAB], cnti[KLAB];
        float sval[KLAB];
        bool present[KLAB], valid[KLAB];
        int nv = 0, ninst = 0;
        for (int k = 0; k < KLAB; ++k) {
            cntk[k] = ws->cnt_k[b][k];
            cnti[k] = ws->cnt_i[b][k];
            sval[k] = ws->sum_val[b][k];
            present[k] = cntk[k] > 0u;
            valid[k]   = present[k] && (k != 0);
            nv += valid[k] ? 1 : 0;
            ninst += present[k] ? 1 : 0;
        }
        float l_agg = 0.f;
        for (int k = 0; k < KLAB; ++k)
            if (valid[k]) l_agg += sval[k] / fmaxf((float)cnti[k], 1.f);
        l_agg /= (float)(nv > 1 ? nv : 1);

        float l_dis = 0.f;
        if (nv >= 2) {
            int pc = 0;
            for (int i = 0; i < KLAB; ++i)
                for (int j = 0; j < KLAB; ++j)
                    if (i != j && valid[i] && valid[j]) {
                        float sq = gram[i][i] + gram[j][j] - 2.f * gram[i][j];
                        float pd = sqrtf(sq > 0.f ? sq : 1.f);
                        float m2 = fmaxf(2.f * DELTA_D - pd, 0.f);
                        l_dis += logf(m2 * m2 + 1.f);
                        ++pc;
                    }
            l_dis /= (float)(pc > 1 ? pc : 1);
        }
        float l_reg = 0.f;
        for (int k = 0; k < KLAB; ++k)
            if (present[k]) l_reg += logf(sqrtf(fmaxf(gram[k][k], 0.f)) + 1.f);
        l_reg = l_reg / (float)(ninst > 1 ? ninst : 1) * 0.001f;

        float l_emb = (nv > 0) ? (l_agg + l_dis + l_reg) : 0.f;

        float A1 = ws->dice[b][0], B1 = ws->dice[b][1], C1 = ws->dice[b][2];
        float A2 = ws->dice[b][3], B2 = ws->dice[b][4], C2 = ws->dice[b][5];
        float lt = 1.f - 2.f * A1 / (B1 + C1 + 2.f * EPS_DICE);
        float lk = 1.f - 2.f * A2 / (B2 + C2 + 2.f * EPS_DICE);

        float total = 1.0f * lt + 0.5f * lk + 0.25f * l_emb;
        atomicAdd(out, total * (1.f / (float)B_));
    }
}

// ---------------- launcher ----------------
extern "C" void kernel_launch(void* const* d_in, const int* in_sizes, int n_in,
                              void* d_out, int out_size, void* d_ws, size_t ws_size,
                              hipStream_t stream) {
    (void)in_sizes; (void)n_in; (void)out_size; (void)ws_size;
    const float* maps        = (const float*)d_in[0];
    const float* gt_texts    = (const float*)d_in[1];
    const float* gt_kernels  = (const float*)d_in[2];
    const int*   gt_instance = (const int*)d_in[3];
    const float* tmask       = (const float*)d_in[4];
    float* out = (float*)d_out;
    Ws* ws = (Ws*)d_ws;

    k_init   <<<dim3(256), dim3(256), 0, stream>>>(ws, out);
    k_negmult<<<dim3((NCELL + 255) / 256, B_), dim3(256), 0, stream>>>(gt_texts, ws);
    k_pass1  <<<dim3(HLO, B_), dim3(256), 0, stream>>>(maps, gt_texts, gt_kernels, gt_instance, tmask, ws);
    k_prep   <<<1, 32, 0, stream>>>(ws);
    for (int p = 0; p < 4; ++p) {
        k_hist  <<<dim3(32, B_), dim3(256), 0, stream>>>(maps, ws, p);
        k_select<<<dim3(B_), dim3(256), 0, stream>>>(ws, p);
    }
    k_pass2 <<<dim3(HLO, B_), dim3(256), 0, stream>>>(maps, gt_texts, gt_kernels, gt_instance, tmask, ws);
    k_final <<<dim3(B_), dim3(32), 0, stream>>>(ws, out);
}